// GCNEncoder_71562745086177
// MI455X (gfx1250) — compile-verified
//
#include <hip/hip_runtime.h>

// CDNA5 / gfx1250 GCN encoder.
// f32 WMMA (V_WMMA_F32_16X16X4_F32) for both GEMMs; wave-per-edge coalesced
// gather + global_atomic_add_f32 scatter for aggregation (h/out fit in 192MB L2).

typedef __attribute__((ext_vector_type(2))) float v2f;
typedef __attribute__((ext_vector_type(8))) float v8f;

// ---------------- degree / norm ----------------
__global__ void k_deg_init(float* __restrict__ deg, int n) {
  int i = blockIdx.x * blockDim.x + threadIdx.x;
  if (i < n) deg[i] = 1.0f;  // self-loop
}

__global__ void k_deg_count(const long long* __restrict__ ei, float* __restrict__ deg,
                            int n_edges) {
  int e = blockIdx.x * blockDim.x + threadIdx.x;
  if (e < n_edges) {
    int dst = (int)ei[(size_t)n_edges + e];
    atomicAdd(&deg[dst], 1.0f);
  }
}

__global__ void k_deg_rsqrt(float* __restrict__ deg, int n) {
  int i = blockIdx.x * blockDim.x + threadIdx.x;
  if (i < n) deg[i] = rsqrtf(deg[i]);  // deg >= 1
}

// ---------------- WMMA GEMM: Y[M,N] = act(X[M,128]) @ W[128,N] ----------------
// Wave computes a 16-row M-tile across all N. A/B layouts per CDNA5 ISA 7.12.2:
//   A (16x4 f32): lane L, vgpr r -> X[m0+(L&15)][k + (L>>4)*2 + r]
//   B (4x16 f32): lane L, vgpr r -> W[k + (L>>4)*2 + r][n0+(L&15)]
//   D (16x16 f32): lane L, vgpr j -> Y[m0 + (L>>4)*8 + j][n0+(L&15)]
template <int N, bool RELU>
__global__ __launch_bounds__(256) void k_gemm_wmma(const float* __restrict__ X,
                                                   const float* __restrict__ W,
                                                   float* __restrict__ Y, int M) {
  constexpr int K  = 128;
  constexpr int NT = N / 16;

  __shared__ float sW[K * N];
  for (int i = threadIdx.x; i < K * N; i += 256) sW[i] = W[i];
  __syncthreads();

  const int warp = threadIdx.x >> 5;
  const int lane = threadIdx.x & 31;
  const int r    = lane & 15;
  const int half = lane >> 4;
  const int m0   = (blockIdx.x * 8 + warp) * 16;

  if (m0 >= M) return;  // fully out-of-bounds wave-tile (tail block only)

  int mrow = m0 + r;
  if (mrow > M - 1) mrow = M - 1;  // clamp loads; stores are guarded below
  const float* __restrict__ xrow = X + (size_t)mrow * K;

  v8f acc[NT];
  for (int t = 0; t < NT; ++t)
    for (int j = 0; j < 8; ++j) acc[t][j] = 0.0f;

  for (int k = 0; k < K; k += 4) {
    const int ka = k + half * 2;
    v2f a;
    a.x = xrow[ka];
    a.y = xrow[ka + 1];
    if (RELU) {
      a.x = fmaxf(a.x, 0.0f);
      a.y = fmaxf(a.y, 0.0f);
    }
#pragma unroll
    for (int t = 0; t < NT; ++t) {
      const int col = t * 16 + r;
      v2f b;
      b.x = sW[ka * N + col];
      b.y = sW[(ka + 1) * N + col];
      acc[t] = __builtin_amdgcn_wmma_f32_16x16x4_f32(false, a, false, b,
                                                     (short)0, acc[t], false, false);
    }
  }

  const int mbase = m0 + half * 8;
  float* __restrict__ yrow = Y + (size_t)mbase * N + r;
  if (m0 + 16 <= M) {
    // Whole tile in bounds (the common case: M % 16 == 0): unguarded stores.
#pragma unroll
    for (int t = 0; t < NT; ++t) {
#pragma unroll
      for (int j = 0; j < 8; ++j) {
        yrow[(size_t)j * N + t * 16] = acc[t][j];
      }
    }
  } else {
#pragma unroll
    for (int t = 0; t < NT; ++t) {
#pragma unroll
      for (int j = 0; j < 8; ++j) {
        const int mm = mbase + j;
        if (mm < M) Y[(size_t)mm * N + t * 16 + r] = acc[t][j];
      }
    }
  }
}

// ---------------- out[i,c] = bias[c] + h[i,c] * dis[i]^2  (self-loop term) ----------------
template <int C>
__global__ void k_init_out(const float* __restrict__ h, const float* __restrict__ bias,
                           const float* __restrict__ dis, float* __restrict__ out, int M) {
  int i = blockIdx.x * blockDim.x + threadIdx.x;
  if (i >= M * C) return;
  const int row = i / C;        // C is a power of two -> shift
  const int c   = i - row * C;
  const float d = dis[row];
  out[i] = fmaf(h[i], d * d, bias[c]);
}

// ---------------- edge scatter: out[dst] += h[src] * dis[src]*dis[dst] ----------------
// One wave32 per edge: 512B (C=128) / 256B (C=64) coalesced gather, contiguous f32 atomics.
template <int C>
__global__ __launch_bounds__(256) void k_edge_scatter(const long long* __restrict__ ei,
                                                      const float* __restrict__ h,
                                                      const float* __restrict__ dis,
                                                      float* __restrict__ out, int n_edges) {
  const int e    = blockIdx.x * 8 + (threadIdx.x >> 5);
  const int lane = threadIdx.x & 31;
  if (e >= n_edges) return;
  const int src = (int)ei[e];
  const int dst = (int)ei[(size_t)n_edges + e];
  const float nm = dis[src] * dis[dst];
  if (C == 128) {
    const float4 v = ((const float4*)(h + (size_t)src * 128))[lane];
    float* o = out + (size_t)dst * 128 + lane * 4;
    atomicAdd(o + 0, v.x * nm);
    atomicAdd(o + 1, v.y * nm);
    atomicAdd(o + 2, v.z * nm);
    atomicAdd(o + 3, v.w * nm);
  } else {
    const float2 v = ((const float2*)(h + (size_t)src * 64))[lane];
    float* o = out + (size_t)dst * 64 + lane * 2;
    atomicAdd(o + 0, v.x * nm);
    atomicAdd(o + 1, v.y * nm);
  }
}

extern "C" void kernel_launch(void* const* d_in, const int* in_sizes, int n_in,
                              void* d_out, int out_size, void* d_ws, size_t ws_size,
                              hipStream_t stream) {
  const float*     x  = (const float*)d_in[0];
  const long long* ei = (const long long*)d_in[1];  // int64 [2, E]
  const float*     W1 = (const float*)d_in[2];      // [128,128]
  const float*     b1 = (const float*)d_in[3];      // [128]
  const float*     W2 = (const float*)d_in[4];      // [128,64]
  const float*     b2 = (const float*)d_in[5];      // [64]
  float* out = (float*)d_out;                       // [M,64] f32

  const int M = in_sizes[0] / 128;   // 100000
  const int E = in_sizes[1] / 2;     // 1600000

  // Workspace carve: dis[M] | hbuf[M*128] (h1, then reused for h2) | t1[M*128] (out1)
  char*  ws   = (char*)d_ws;
  float* dis  = (float*)ws;
  size_t off  = (((size_t)M * 4) + 511) & ~(size_t)511;
  float* hbuf = (float*)(ws + off);
  float* t1   = hbuf + (size_t)M * 128;

  const int TB = 256;

  // degree -> dis = rsqrt(deg)
  k_deg_init <<<(M + TB - 1) / TB, TB, 0, stream>>>(dis, M);
  k_deg_count<<<(E + TB - 1) / TB, TB, 0, stream>>>(ei, dis, E);
  k_deg_rsqrt<<<(M + TB - 1) / TB, TB, 0, stream>>>(dis, M);

  const int gemm_blocks = (M + 127) / 128;

  // layer 1: h1 = x @ W1 ; out1 = b1 + dis^2*h1 + scatter(norm*h1)
  k_gemm_wmma<128, false><<<gemm_blocks, TB, 0, stream>>>(x, W1, hbuf, M);
  k_init_out<128><<<(M * 128 + TB - 1) / TB, TB, 0, stream>>>(hbuf, b1, dis, t1, M);
  k_edge_scatter<128><<<(E + 7) / 8, TB, 0, stream>>>(ei, hbuf, dis, t1, E);

  // layer 2: h2 = relu(out1) @ W2 ; out = b2 + dis^2*h2 + scatter(norm*h2)
  k_gemm_wmma<64, true><<<gemm_blocks, TB, 0, stream>>>(t1, W2, hbuf, M);
  k_init_out<64><<<(M * 64 + TB - 1) / TB, TB, 0, stream>>>(hbuf, b2, dis, out, M);
  k_edge_scatter<64><<<(E + 7) / 8, TB, 0, stream>>>(ei, hbuf, dis, out, E);
}